// IsoformScorer_34772055229164
// MI455X (gfx1250) — compile-verified
//
#include <hip/hip_runtime.h>
#include <climits>

#define ALPHA 0.5f
#define BETA  0.3f
#define GAMMA 0.6f

#define CAND_CAP 4096
#define HIT_CAP  8192
#define KMAX     128
#define NPAIR_CAP 512
#define SCAN_BLOCK 1024

typedef __attribute__((ext_vector_type(2))) float v2f;
typedef __attribute__((ext_vector_type(8))) float v8f;

__device__ __forceinline__ float sigmoidf_(float x) { return 1.0f / (1.0f + expf(-x)); }

// ---------------- init: zero histogram / counters / pair scores ----------------
__global__ void k_init(unsigned* hist, int* counters, float* pair_scores) {
  int i = blockIdx.x * blockDim.x + threadIdx.x;
  if (i < 65536) hist[i] = 0u;
  if (i < NPAIR_CAP) pair_scores[i] = -1e30f;
  if (i < 4) counters[i] = 0;
}

// ---------------- start probs + 16-bit float-bit histogram ----------------
__global__ void k_probs_hist(const float* __restrict__ start_l, const int* __restrict__ tx,
                             float* __restrict__ sprob, unsigned* __restrict__ hist, int T) {
  int t = blockIdx.x * blockDim.x + threadIdx.x;
  if (t >= T) return;
  float p = sigmoidf_(start_l[tx[t]]);   // p in [0,1]: positive floats order like their bits
  sprob[t] = p;
  unsigned bin = __float_as_uint(p) >> 16;
  atomicAdd(&hist[bin], 1u);
}

// ---------------- find threshold bin: max bin B with count(bin >= B) >= K ----------------
__global__ void k_thresh(const unsigned* __restrict__ hist, int* counters,
                         const int* __restrict__ topk_p) {
  __shared__ unsigned partial[1024];
  int K = topk_p[0]; if (K < 1) K = 1; if (K > KMAX) K = KMAX;
  unsigned s = 0;
  int base = threadIdx.x * 64;
  for (int i = 0; i < 64; ++i) s += hist[base + i];
  partial[threadIdx.x] = s;
  __syncthreads();
  if (threadIdx.x == 0) {
    unsigned suffix = 0; int coarse = -1;
    for (int b = 1023; b >= 0; --b) {
      if (suffix + partial[b] >= (unsigned)K) { coarse = b; break; }
      suffix += partial[b];
    }
    int thresh = 0;
    if (coarse >= 0) {
      for (int i = 63; i >= 0; --i) {
        unsigned h = hist[coarse * 64 + i];
        if (suffix + h >= (unsigned)K) { thresh = coarse * 64 + i; break; }
        suffix += h;
      }
    }
    counters[2] = thresh;
  }
}

// ---------------- collect candidates at/above threshold bin ----------------
__global__ void k_collect(const float* __restrict__ sprob, int* counters,
                          float* __restrict__ cand_val, int* __restrict__ cand_idx, int T) {
  int t = blockIdx.x * blockDim.x + threadIdx.x;
  if (t >= T) return;
  float p = sprob[t];
  int bin = (int)(__float_as_uint(p) >> 16);
  if (bin >= counters[2]) {
    int slot = atomicAdd(&counters[0], 1);
    if (slot < CAND_CAP) { cand_val[slot] = p; cand_idx[slot] = t; }
  }
}

// ---------------- exact top-K among candidates (value desc, index asc = lax.top_k set) ----------
__global__ void k_select(float* cand_val, const int* __restrict__ cand_idx, const int* counters,
                         float* __restrict__ sel_val, int* __restrict__ sel_idx,
                         const int* __restrict__ topk_p) {
  int K = topk_p[0]; if (K < 1) K = 1; if (K > KMAX) K = KMAX;
  int n = counters[0]; if (n > CAND_CAP) n = CAND_CAP;
  __shared__ float bv[256]; __shared__ int bi[256]; __shared__ int bs[256];
  int tid = threadIdx.x;
  for (int k = 0; k < K; ++k) {
    float best = -3.0f; int bidx = INT_MAX; int bslot = -1;
    for (int s = tid; s < n; s += blockDim.x) {
      float v = cand_val[s];
      if (v < -1.0f) continue;                 // taken marker
      int id = cand_idx[s];
      if (v > best || (v == best && id < bidx)) { best = v; bidx = id; bslot = s; }
    }
    bv[tid] = best; bi[tid] = bidx; bs[tid] = bslot;
    __syncthreads();
    for (int off = blockDim.x / 2; off > 0; off >>= 1) {
      if (tid < off) {
        if (bv[tid + off] > bv[tid] || (bv[tid + off] == bv[tid] && bi[tid + off] < bi[tid])) {
          bv[tid] = bv[tid + off]; bi[tid] = bi[tid + off]; bs[tid] = bs[tid + off];
        }
      }
      __syncthreads();
    }
    if (tid == 0) {
      if (bs[0] >= 0) { sel_val[k] = bv[0]; sel_idx[k] = bi[0]; cand_val[bs[0]] = -2.0f; }
      else            { sel_val[k] = -1e30f; sel_idx[k] = 0; }
    }
    __syncthreads();
  }
}

// ---------------- global stop-hit list: t with stop_logit[g(t)] > 0 (sigmoid > 0.5) ----------
__global__ void k_hits(const float* __restrict__ stop_l, const int* __restrict__ tx,
                       int* counters, int* __restrict__ hits, int T) {
  int t = blockIdx.x * blockDim.x + threadIdx.x;
  if (t >= T) return;
  if (stop_l[tx[t]] > 0.0f) {
    int slot = atomicAdd(&counters[1], 1);
    if (slot < HIT_CAP) hits[slot] = t;
  }
}

// ---------------- 9 inclusive prefix sums PS[f][r][i] of softmax(frame[g(r+3i)])[f] ----------
__global__ __launch_bounds__(SCAN_BLOCK) void k_scan(const float* __restrict__ frame_l,
                                                     const int* __restrict__ tx,
                                                     float* __restrict__ ps, int T, int Npad) {
  int a = blockIdx.x;            // a = f*3 + r
  int f = a / 3, r = a % 3;
  int N = (T - r + 2) / 3;       // ceil((T-r)/3)
  __shared__ float wsum[32];
  __shared__ float carry_s;
  if (threadIdx.x == 0) carry_s = 0.0f;
  __syncthreads();
  int lane = threadIdx.x & 31, warp = threadIdx.x >> 5;
  float* out = ps + (size_t)a * Npad;
  for (int base = 0; base < N; base += SCAN_BLOCK) {
    int i = base + threadIdx.x;
    float v = 0.0f;
    if (i < N) {
      int t = r + 3 * i;                      // t < T guaranteed
      int g = tx[t];
      float a0 = frame_l[3 * g + 0], a1 = frame_l[3 * g + 1], a2 = frame_l[3 * g + 2];
      float m = fmaxf(a0, fmaxf(a1, a2));
      float e0 = expf(a0 - m), e1 = expf(a1 - m), e2 = expf(a2 - m);
      float sum = e0 + e1 + e2;
      v = ((f == 0) ? e0 : (f == 1) ? e1 : e2) / sum;
    }
    // wave32 inclusive scan
    float x = v;
    for (int off = 1; off < 32; off <<= 1) {
      float y = __shfl_up(x, off, 32);
      if (lane >= off) x += y;
    }
    if (lane == 31) wsum[warp] = x;
    __syncthreads();
    if (warp == 0) {
      float w = wsum[lane];
      for (int off = 1; off < 32; off <<= 1) {
        float y = __shfl_up(w, off, 32);
        if (lane >= off) w += y;
      }
      wsum[lane] = w;
    }
    __syncthreads();
    float add = (warp > 0) ? wsum[warp - 1] : 0.0f;
    float incl = x + add + carry_s;
    if (i < N) out[i] = incl;
    __syncthreads();
    if (threadIdx.x == SCAN_BLOCK - 1) carry_s = incl;  // running total carries to next tile
    __syncthreads();
  }
}

// ---------------- per-(start,frame) pair score, O(#hits) each ----------------
__global__ void k_pairs(float* __restrict__ pair_scores,
                        const float* __restrict__ sel_val, const int* __restrict__ sel_idx,
                        const int* __restrict__ hits, const int* __restrict__ counters,
                        const float* __restrict__ ps, const int* __restrict__ tx,
                        const float* __restrict__ stop_l,
                        const int* __restrict__ topk_p, const int* __restrict__ lastj_p,
                        const int* __restrict__ nex_p, int T, int Npad) {
  int p = blockIdx.x * blockDim.x + threadIdx.x;
  if (p >= NPAIR_CAP) return;
  int K = topk_p[0]; if (K < 1) K = 1; if (K > KMAX) K = KMAX;
  int np = K * 3; if (np > NPAIR_CAP) np = NPAIR_CAP;
  if (p >= np) { pair_scores[p] = -1e30f; return; }
  int k = p / 3, f = p % 3;
  float sp = sel_val[k];
  int srel = sel_idx[k];
  int t0 = srel + f;                      // first codon position; may be >= T (then no hit possible)
  int nh = counters[1]; if (nh > HIT_CAP) nh = HIT_CAP;
  int tstar = INT_MAX;
  for (int h = 0; h < nh; ++h) {
    int t = hits[h];
    if (t >= t0 && ((t - t0) % 3) == 0 && t < tstar) tstar = t;  // hits all have t < T (valid)
  }
  float sc;
  if (tstar != INT_MAX) {
    int r = t0 % 3;
    int i0 = t0 / 3;
    int istar = tstar / 3;                // tstar ≡ r (mod 3)
    const float* psA = ps + (size_t)(f * 3 + r) * Npad;
    float seg = psA[istar] - (i0 > 0 ? psA[i0 - 1] : 0.0f);
    float mean_f = seg / (float)(istar - i0 + 1);
    float stop_p = sigmoidf_(stop_l[tx[tstar]]);
    sc = ALPHA * sp + BETA * mean_f + ALPHA * stop_p;
    if (nex_p[0] > 1 && tstar < lastj_p[0] - 55) sc -= GAMMA;
  } else {
    sc = ALPHA * sp - GAMMA;
  }
  pair_scores[p] = sc;
}

// ---------------- finalize on one full wave32 (EXEC all-1s for WMMA) ----------------
__global__ __launch_bounds__(32) void k_final(
    const float* __restrict__ donor, const float* __restrict__ acceptor,
    const float* __restrict__ tss_l, const float* __restrict__ polya_l,
    const int* __restrict__ didx, int nd, const int* __restrict__ aidx, int na,
    const float* __restrict__ pair_scores, float* __restrict__ out,
    const int* __restrict__ tss_i, const int* __restrict__ pa_i, const int* __restrict__ nex_p,
    const float* __restrict__ w_spl, const float* __restrict__ w_tss,
    const float* __restrict__ w_pa, const float* __restrict__ w_orf,
    const float* __restrict__ w_len, int L) {
  int lane = threadIdx.x;
  int nj = nd + na;

  // Per-lane partial sums of the gathered junction logits (2 A-matrix slots per lane = 64 slots)
  float acc0 = 0.0f, acc1 = 0.0f;
  for (int e = lane; e < nj; e += 64)
    acc0 += (e < nd) ? donor[didx[e]] : acceptor[aidx[e - nd]];
  for (int e = lane + 32; e < nj; e += 64)
    acc1 += (e < nd) ? donor[didx[e]] : acceptor[aidx[e - nd]];

  // Dot-with-ones via V_WMMA_F32_16X16X4_F32: D[m][n] = rowsum(A[m]) for all n
  v2f A; A.x = acc0; A.y = acc1;
  v2f B; B.x = 1.0f; B.y = 1.0f;
  v8f C = {};
  v8f D = __builtin_amdgcn_wmma_f32_16x16x4_f32(false, A, false, B, (short)0, C, false, false);
  float part = D[0] + D[1] + D[2] + D[3] + D[4] + D[5] + D[6] + D[7]; // rows 0-7 (lanes<16) / 8-15
  float tot = part + __shfl_xor(part, 16, 32);                        // full Σ A on every lane
  float s_spl = tot / (float)nj;

  // s_tss / s_pa (window max of log(sigmoid+1e-9)); computed uniformly by all lanes
  float s_tss = -1e30f;
  {
    int c = tss_i[0];
    int lo = c - 1 > 0 ? c - 1 : 0;
    int hi = c + 2 < L ? c + 2 : L;
    for (int t = lo; t < hi; ++t)
      s_tss = fmaxf(s_tss, logf(sigmoidf_(tss_l[t]) + 1e-9f));
  }
  float s_pa = -1e30f;
  {
    int c = pa_i[0];
    int lo = c - 1 > 0 ? c - 1 : 0;
    int hi = c + 2 < L ? c + 2 : L;
    for (int t = lo; t < hi; ++t)
      s_pa = fmaxf(s_pa, logf(sigmoidf_(polya_l[t]) + 1e-9f));
  }

  // s_orf = max(max pair score, 0)
  float m = -1e30f;
  for (int s = lane; s < NPAIR_CAP; s += 32) m = fmaxf(m, pair_scores[s]);
  for (int off = 16; off > 0; off >>= 1) m = fmaxf(m, __shfl_xor(m, off, 32));
  float s_orf = fmaxf(m, 0.0f);

  int E = nex_p[0];
  float s_len = (E < 2 || E > 20) ? -0.5f : 0.0f;

  if (lane == 0)
    out[0] = w_spl[0] * s_spl + w_tss[0] * s_tss + w_pa[0] * s_pa +
             w_orf[0] * s_orf + w_len[0] * s_len;
}

extern "C" void kernel_launch(void* const* d_in, const int* in_sizes, int n_in,
                              void* d_out, int out_size, void* d_ws, size_t ws_size,
                              hipStream_t stream) {
  (void)n_in; (void)out_size; (void)ws_size;
  const float* donor    = (const float*)d_in[0];
  const float* acceptor = (const float*)d_in[1];
  const float* start_l  = (const float*)d_in[2];
  const float* stop_l   = (const float*)d_in[3];
  const float* frame_l  = (const float*)d_in[4];
  const float* tss_l    = (const float*)d_in[5];
  const float* polya_l  = (const float*)d_in[6];
  const float* w_spl    = (const float*)d_in[7];
  const float* w_tss    = (const float*)d_in[8];
  const float* w_pa     = (const float*)d_in[9];
  const float* w_orf    = (const float*)d_in[10];
  const float* w_len    = (const float*)d_in[11];
  const int*   tx       = (const int*)d_in[12];  int T  = in_sizes[12];
  const int*   didx     = (const int*)d_in[13];  int nd = in_sizes[13];
  const int*   aidx     = (const int*)d_in[14];  int na = in_sizes[14];
  const int*   tss_i    = (const int*)d_in[15];
  const int*   pa_i     = (const int*)d_in[16];
  const int*   lastj    = (const int*)d_in[17];
  const int*   nex      = (const int*)d_in[18];
  const int*   topk     = (const int*)d_in[19];
  int L = in_sizes[0];
  int N = (T + 2) / 3;
  int Npad = ((N + 255) / 256) * 256;

  size_t off = 0;
  auto carve = [&](size_t bytes) -> char* {
    char* p = (char*)d_ws + off;
    off += (bytes + 255) & ~(size_t)255;
    return p;
  };
  float*    sprob       = (float*)carve((size_t)T * 4);
  unsigned* hist        = (unsigned*)carve(65536 * 4);
  float*    cand_val    = (float*)carve(CAND_CAP * 4);
  int*      cand_idx    = (int*)carve(CAND_CAP * 4);
  int*      counters    = (int*)carve(64);
  float*    sel_val     = (float*)carve(KMAX * 4);
  int*      sel_idx     = (int*)carve(KMAX * 4);
  int*      hits        = (int*)carve(HIT_CAP * 4);
  float*    ps          = (float*)carve((size_t)9 * Npad * 4);
  float*    pair_scores = (float*)carve(NPAIR_CAP * 4);
  float*    out         = (float*)d_out;

  const int tb = 256;
  k_init<<<(65536 + tb - 1) / tb, tb, 0, stream>>>(hist, counters, pair_scores);
  k_probs_hist<<<(T + tb - 1) / tb, tb, 0, stream>>>(start_l, tx, sprob, hist, T);
  k_thresh<<<1, 1024, 0, stream>>>(hist, counters, topk);
  k_collect<<<(T + tb - 1) / tb, tb, 0, stream>>>(sprob, counters, cand_val, cand_idx, T);
  k_select<<<1, 256, 0, stream>>>(cand_val, cand_idx, counters, sel_val, sel_idx, topk);
  k_hits<<<(T + tb - 1) / tb, tb, 0, stream>>>(stop_l, tx, counters, hits, T);
  k_scan<<<9, SCAN_BLOCK, 0, stream>>>(frame_l, tx, ps, T, Npad);
  k_pairs<<<(NPAIR_CAP + tb - 1) / tb, tb, 0, stream>>>(pair_scores, sel_val, sel_idx, hits,
                                                        counters, ps, tx, stop_l, topk, lastj,
                                                        nex, T, Npad);
  k_final<<<1, 32, 0, stream>>>(donor, acceptor, tss_l, polya_l, didx, nd, aidx, na,
                                pair_scores, out, tss_i, pa_i, nex,
                                w_spl, w_tss, w_pa, w_orf, w_len, L);
}